// BigGNN_46694884442485
// MI455X (gfx1250) — compile-verified
//
#include <hip/hip_runtime.h>
#include <cstddef>

// BigGNN on gfx1250: fp32 WMMA (v_wmma_f32_16x16x4_f32) GEMMs + sparse edge
// softmax + dense cross-attention (GEMM/softmax/GEMM) + MLP.
//
// Input flattening assumption (insertion order, recursive):
//  0 x_1(384*300) 1 x_2 2 edge_idx_1(2*E) 3 edge_idx_2 4 edge_attr_1(E*300)
//  5 edge_attr_2
//  6..14  tsa: Wq bq Wk bk Wv bv We Wskip bskip
// 15..23  gsa, 24..32 tca, 33..41 gca
// 42..47  mlp: W1 b1 W2 b2 W3 b3
// Output: [x1p(300) | x2p(300) | sigmoid(1)] = 601 floats.

#define HEADS 2
#define CDIM  300
#define HCDIM 600

typedef __attribute__((ext_vector_type(2))) float v2f;
typedef __attribute__((ext_vector_type(8))) float v8f;

__device__ __forceinline__ float leakyf(float v) { return v >= 0.f ? v : 0.01f * v; }
__device__ __forceinline__ int mini(int a, int b) { return a < b ? a : b; }

// monotonic float<->uint mapping for atomic max
__device__ __forceinline__ unsigned f2ord(float f) {
    unsigned u = __float_as_uint(f);
    return (u & 0x80000000u) ? ~u : (u | 0x80000000u);
}
__device__ __forceinline__ float ord2f(unsigned u) {
    return (u & 0x80000000u) ? __uint_as_float(u ^ 0x80000000u) : __uint_as_float(~u);
}
#define NEG_INF_ORD 0x007FFFFFu   // f2ord(-inf)

// ---------------------------------------------------------------------------
// WMMA fp32 GEMM.  2x2 tiles of 16x16 per wave (wave covers 32x32 of C),
// 4 waves / 128-thread block.  Branchless inner loop: row/col indices are
// CLAMPED instead of masked -- out-of-range A rows / B cols only pollute
// out-of-range C entries, which the store guard drops.
// C = act( alpha * A@B(t) + (beta ? C : 0) + bias )
// A: MxK (lda), B: KxN (ldb) for nn, NxK (ldb) for nt.  K % 4 == 0 required.
// All lda/ldb/ldc and buffer bases are 8-byte aligned -> v2f loads are legal.
// Fragment layouts per CDNA5 ISA 7.12.2:
//   A 16x4 : lane L -> row = L&15, holds K = (L>>4)*2 + {0,1}
//   B 4x16 : lane L -> col = L&15, holds K = (L>>4)*2 + {0,1}
//   C 16x16: vgpr i, lane L -> row = i + (L>>4)*8, col = L&15
// ---------------------------------------------------------------------------
template <int NT>
__global__ __launch_bounds__(128) void k_gemm(const float* __restrict__ A, int lda,
                                              const float* __restrict__ B, int ldb,
                                              float* __restrict__ C, int ldc,
                                              int M, int N, int K, float alpha,
                                              const float* __restrict__ bias,
                                              int beta, int act) {
    const int wave = threadIdx.x >> 5;
    const int lane = threadIdx.x & 31;
    const int tm0 = (blockIdx.y * 4 + wave) * 32;   // row base of this wave's 32x32 block
    const int tn0 = blockIdx.x * 32;                // col base
    if (tm0 >= M) return;                           // wave-uniform exit (EXEC full for WMMA)

    const int r  = lane & 15;
    const int kg = (lane >> 4) * 2;                 // 0 or 2

    // clamped fragment row/col indices (always in-bounds loads)
    const int ar0 = mini(tm0 + r,      M - 1);
    const int ar1 = mini(tm0 + 16 + r, M - 1);
    const int bc0 = mini(tn0 + r,      N - 1);
    const int bc1 = mini(tn0 + 16 + r, N - 1);
    const float* ap0 = A + (size_t)ar0 * lda + kg;
    const float* ap1 = A + (size_t)ar1 * lda + kg;

    v8f acc00 = {0.f,0.f,0.f,0.f,0.f,0.f,0.f,0.f};
    v8f acc01 = acc00, acc10 = acc00, acc11 = acc00;

    for (int k0 = 0; k0 < K; k0 += 4) {
        v2f a0 = *(const v2f*)(ap0 + k0);
        v2f a1 = *(const v2f*)(ap1 + k0);
        v2f b0, b1;
        if (NT) {   // B is N x K: contiguous pair along K
            b0 = *(const v2f*)(B + (size_t)bc0 * ldb + k0 + kg);
            b1 = *(const v2f*)(B + (size_t)bc1 * ldb + k0 + kg);
        } else {    // B is K x N: strided pair
            const int ka = k0 + kg;
            b0.x = B[(size_t)ka * ldb + bc0];
            b0.y = B[(size_t)(ka + 1) * ldb + bc0];
            b1.x = B[(size_t)ka * ldb + bc1];
            b1.y = B[(size_t)(ka + 1) * ldb + bc1];
        }
        acc00 = __builtin_amdgcn_wmma_f32_16x16x4_f32(false, a0, false, b0, (short)0, acc00, false, false);
        acc01 = __builtin_amdgcn_wmma_f32_16x16x4_f32(false, a0, false, b1, (short)0, acc01, false, false);
        acc10 = __builtin_amdgcn_wmma_f32_16x16x4_f32(false, a1, false, b0, (short)0, acc10, false, false);
        acc11 = __builtin_amdgcn_wmma_f32_16x16x4_f32(false, a1, false, b1, (short)0, acc11, false, false);
    }

    const int col0 = tn0 + (lane & 15);
    const int col1 = col0 + 16;
    const int rh   = (lane >> 4) << 3;
#pragma unroll
    for (int i = 0; i < 8; ++i) {
        const int row0 = tm0 + i + rh;
        const int row1 = row0 + 16;
        float v;
#define EMIT(ACC, ROW, COL)                                        \
        if ((ROW) < M && (COL) < N) {                              \
            v = ACC[i] * alpha;                                    \
            if (beta) v += C[(size_t)(ROW) * ldc + (COL)];         \
            if (bias) v += bias[(COL)];                            \
            if (act)  v = leakyf(v);                               \
            C[(size_t)(ROW) * ldc + (COL)] = v;                    \
        }
        EMIT(acc00, row0, col0)
        EMIT(acc01, row0, col1)
        EMIT(acc10, row1, col0)
        EMIT(acc11, row1, col1)
#undef EMIT
    }
}

// ---------------------------------------------------------------------------
// Sparse graph attention helpers
// ---------------------------------------------------------------------------
__global__ void k_fill_f32(float* p, float v, int n) {
    int i = blockIdx.x * blockDim.x + threadIdx.x;
    if (i < n) p[i] = v;
}
__global__ void k_fill_u32(unsigned* p, unsigned v, int n) {
    int i = blockIdx.x * blockDim.x + threadIdx.x;
    if (i < n) p[i] = v;
}

// alpha[e,h] = <q[dst], k[src]+ep[e]> / sqrt(C); atomic segment max
__global__ void k_edge_alpha(const float* __restrict__ Q, const float* __restrict__ Kn,
                             const float* __restrict__ EP, const int* __restrict__ src,
                             const int* __restrict__ dst, float* __restrict__ alpha,
                             unsigned* __restrict__ menc, int E) {
    int t = blockIdx.x * blockDim.x + threadIdx.x;
    if (t >= E * HEADS) return;
    int e = t >> 1, h = t & 1;
    int s = src[e], d = dst[e];
    const float* q  = Q  + (size_t)d * HCDIM + h * CDIM;
    const float* kk = Kn + (size_t)s * HCDIM + h * CDIM;
    const float* ep = EP + (size_t)e * HCDIM + h * CDIM;
    float acc = 0.f;
    for (int c = 0; c < CDIM; ++c) acc += q[c] * (kk[c] + ep[c]);
    acc *= 0.05773502691896258f;   // 1/sqrt(300)
    alpha[t] = acc;
    atomicMax(&menc[d * HEADS + h], f2ord(acc));
}

// a = exp(alpha - m[dst]); atomic segment sum
__global__ void k_edge_exp(const int* __restrict__ dst, float* __restrict__ alpha,
                           const unsigned* __restrict__ menc, float* __restrict__ ssum,
                           int E) {
    int t = blockIdx.x * blockDim.x + threadIdx.x;
    if (t >= E * HEADS) return;
    int e = t >> 1, h = t & 1;
    int d = dst[e];
    float a = expf(alpha[t] - ord2f(menc[d * HEADS + h]));
    alpha[t] = a;
    atomicAdd(&ssum[d * HEADS + h], a);
}

// w = a / (s[dst] + 1e-16)
__global__ void k_edge_weight(const int* __restrict__ dst, float* __restrict__ alpha,
                              const float* __restrict__ ssum, int E) {
    int t = blockIdx.x * blockDim.x + threadIdx.x;
    if (t >= E * HEADS) return;
    int e = t >> 1, h = t & 1;
    alpha[t] = alpha[t] / (ssum[dst[e] * HEADS + h] + 1e-16f);
}

// Out[dst, c] += 0.5 * w[e,h] * (V[src, h*C+c] + EP[e, h*C+c])   (head mean folded)
__global__ void k_edge_aggr(const int* __restrict__ src, const int* __restrict__ dst,
                            const float* __restrict__ alpha, const float* __restrict__ V,
                            const float* __restrict__ EP, float* __restrict__ Out, int E) {
    int t = blockIdx.x * blockDim.x + threadIdx.x;
    if (t >= E * HCDIM) return;
    int e = t / HCDIM, j = t - e * HCDIM;
    int h = (j >= CDIM);
    int d = dst[e], s = src[e];
    float w = alpha[e * HEADS + h];
    float contrib = 0.5f * w * (V[(size_t)s * HCDIM + j] + EP[(size_t)e * HCDIM + j]);
    atomicAdd(&Out[(size_t)d * CDIM + (j - h * CDIM)], contrib);
}

__global__ void k_leaky(float* p, int n) {
    int i = blockIdx.x * blockDim.x + threadIdx.x;
    if (i < n) p[i] = leakyf(p[i]);
}

// ---------------------------------------------------------------------------
// Dense cross-attention helpers
// ---------------------------------------------------------------------------
// combined bias: B2K = bk + colsum(We), B2V = bv + colsum(We)  (edge_attr == 1)
__global__ void k_cross_bias(const float* __restrict__ We, const float* __restrict__ bk,
                             const float* __restrict__ bv, float* __restrict__ B2K,
                             float* __restrict__ B2V) {
    int j = blockIdx.x * blockDim.x + threadIdx.x;
    if (j >= HCDIM) return;
    float s = 0.f;
    for (int i = 0; i < CDIM; ++i) s += We[(size_t)i * HCDIM + j];
    B2K[j] = bk[j] + s;
    B2V[j] = bv[j] + s;
}

// row softmax with +1e-16 denominator; one block per row
__global__ __launch_bounds__(128) void k_softmax_row(float* __restrict__ S, int width) {
    float* row = S + (size_t)blockIdx.x * width;
    __shared__ float red[128];
    float m = -INFINITY;
    for (int j = threadIdx.x; j < width; j += 128) m = fmaxf(m, row[j]);
    red[threadIdx.x] = m; __syncthreads();
    for (int s = 64; s > 0; s >>= 1) {
        if (threadIdx.x < s) red[threadIdx.x] = fmaxf(red[threadIdx.x], red[threadIdx.x + s]);
        __syncthreads();
    }
    m = red[0]; __syncthreads();
    float sum = 0.f;
    for (int j = threadIdx.x; j < width; j += 128) {
        float e = expf(row[j] - m);
        row[j] = e; sum += e;
    }
    red[threadIdx.x] = sum; __syncthreads();
    for (int s = 64; s > 0; s >>= 1) {
        if (threadIdx.x < s) red[threadIdx.x] += red[threadIdx.x + s];
        __syncthreads();
    }
    float inv = 1.f / (red[0] + 1e-16f);
    for (int j = threadIdx.x; j < width; j += 128) row[j] *= inv;
}

// ---------------------------------------------------------------------------
// Pooling + MLP
// ---------------------------------------------------------------------------
__global__ void k_meancols(const float* __restrict__ X, int n, float* __restrict__ out) {
    int c = blockIdx.x * blockDim.x + threadIdx.x;
    if (c >= CDIM) return;
    float s = 0.f;
    for (int i = 0; i < n; ++i) s += X[(size_t)i * CDIM + c];
    out[c] = s / (float)n;
}

__global__ void k_vecmat(const float* __restrict__ h, const float* __restrict__ W,
                         const float* __restrict__ b, float* __restrict__ y,
                         int In, int N, int act) {
    int j = blockIdx.x * blockDim.x + threadIdx.x;
    if (j >= N) return;
    float s = b[j];
    for (int i = 0; i < In; ++i) s += h[i] * W[(size_t)i * N + j];
    if (act) s = leakyf(s);
    y[j] = s;
}

__global__ __launch_bounds__(128) void k_final(const float* __restrict__ y2,
                                               const float* __restrict__ W3,
                                               const float* __restrict__ b3,
                                               float* __restrict__ out) {
    __shared__ float red[128];
    float s = 0.f;
    for (int i = threadIdx.x; i < CDIM; i += 128) s += y2[i] * W3[i];
    red[threadIdx.x] = s; __syncthreads();
    for (int k = 64; k > 0; k >>= 1) {
        if (threadIdx.x < k) red[threadIdx.x] += red[threadIdx.x + k];
        __syncthreads();
    }
    if (threadIdx.x == 0) {
        float v = red[0] + b3[0];
        out[0] = 1.f / (1.f + expf(-v));
    }
}

// ---------------------------------------------------------------------------
// Host side
// ---------------------------------------------------------------------------
static void gemm(hipStream_t st, int nt, const float* A, int lda, const float* B, int ldb,
                 float* C, int ldc, int M, int N, int K, float alpha,
                 const float* bias, int beta, int act) {
    dim3 block(128);
    int wm = (M + 31) / 32;        // 32 rows per wave
    int wn = (N + 31) / 32;        // 32 cols per wave
    dim3 grid(wn, (wm + 3) / 4);   // 4 waves per block stacked along M
    if (nt)
        k_gemm<1><<<grid, block, 0, st>>>(A, lda, B, ldb, C, ldc, M, N, K, alpha, bias, beta, act);
    else
        k_gemm<0><<<grid, block, 0, st>>>(A, lda, B, ldb, C, ldc, M, N, K, alpha, bias, beta, act);
}

struct TP { const float *Wq, *bq, *Wk, *bk, *Wv, *bv, *We, *Wskip, *bskip; };
static TP getTP(void* const* d_in, int base) {
    TP p;
    p.Wq = (const float*)d_in[base + 0]; p.bq = (const float*)d_in[base + 1];
    p.Wk = (const float*)d_in[base + 2]; p.bk = (const float*)d_in[base + 3];
    p.Wv = (const float*)d_in[base + 4]; p.bv = (const float*)d_in[base + 5];
    p.We = (const float*)d_in[base + 6];
    p.Wskip = (const float*)d_in[base + 7]; p.bskip = (const float*)d_in[base + 8];
    return p;
}

extern "C" void kernel_launch(void* const* d_in, const int* in_sizes, int n_in,
                              void* d_out, int out_size, void* d_ws, size_t ws_size,
                              hipStream_t stream) {
    const float* x1  = (const float*)d_in[0];
    const float* x2  = (const float*)d_in[1];
    const int*   ei1 = (const int*)d_in[2];
    const int*   ei2 = (const int*)d_in[3];
    const float* ea1 = (const float*)d_in[4];
    const float* ea2 = (const float*)d_in[5];
    const TP tsa = getTP(d_in, 6), gsa = getTP(d_in, 15);
    const TP tca = getTP(d_in, 24), gca = getTP(d_in, 33);
    const float* W1 = (const float*)d_in[42]; const float* b1 = (const float*)d_in[43];
    const float* W2 = (const float*)d_in[44]; const float* b2 = (const float*)d_in[45];
    const float* W3 = (const float*)d_in[46]; const float* b3 = (const float*)d_in[47];

    const int n1 = in_sizes[0] / CDIM;
    const int n2 = in_sizes[1] / CDIM;
    const int E1 = in_sizes[2] / 2;
    const int E2 = in_sizes[3] / 2;
    const int nmax = n1 > n2 ? n1 : n2;
    const int Emax = E1 > E2 ? E1 : E2;

    // workspace carve-up (floats, 256B aligned blocks)
    float* ws = (float*)d_ws;
    size_t off = 0;
    auto alloc = [&](size_t n) { float* p = ws + off; off += (n + 63) & ~(size_t)63; return p; };
    float* Q    = alloc((size_t)nmax * HCDIM);
    float* Kb   = alloc((size_t)nmax * HCDIM);
    float* Vb   = alloc((size_t)nmax * HCDIM);
    float* EP   = alloc((size_t)Emax * HCDIM);
    float* ALP  = alloc((size_t)Emax * HEADS);
    float* MEN  = alloc((size_t)nmax * HEADS);   // used as unsigned
    float* SSUM = alloc((size_t)nmax * HEADS);
    float* X1P  = alloc((size_t)n1 * CDIM);
    float* X2P  = alloc((size_t)n2 * CDIM);
    float* X1PP = alloc((size_t)n1 * CDIM);
    float* X2PP = alloc((size_t)n2 * CDIM);
    float* S    = alloc((size_t)HEADS * nmax * nmax);
    float* B2K  = alloc(HCDIM);
    float* B2V  = alloc(HCDIM);
    float* Y1   = alloc(HCDIM);
    float* Y2   = alloc(CDIM);
    (void)ws_size; (void)n_in; (void)out_size;

    auto cdiv = [](int a, int b) { return (a + b - 1) / b; };
    const float invSqrtC = 0.05773502691896258f;

    // ---- sparse self-attention conv ----
    auto self_attn = [&](const float* x, const int* ei, const float* ea,
                         const TP& p, int n, int E, float* Xout) {
        const int* src = ei;
        const int* dst = ei + E;
        gemm(stream, 0, x, CDIM, p.Wq, HCDIM, Q,  HCDIM, n, HCDIM, CDIM, 1.f, p.bq, 0, 0);
        gemm(stream, 0, x, CDIM, p.Wk, HCDIM, Kb, HCDIM, n, HCDIM, CDIM, 1.f, p.bk, 0, 0);
        gemm(stream, 0, x, CDIM, p.Wv, HCDIM, Vb, HCDIM, n, HCDIM, CDIM, 1.f, p.bv, 0, 0);
        gemm(stream, 0, ea, CDIM, p.We, HCDIM, EP, HCDIM, E, HCDIM, CDIM, 1.f, nullptr, 0, 0);
        gemm(stream, 0, x, CDIM, p.Wskip, CDIM, Xout, CDIM, n, CDIM, CDIM, 1.f, p.bskip, 0, 0);
        k_fill_u32<<<cdiv(n * HEADS, 256), 256, 0, stream>>>((unsigned*)MEN, NEG_INF_ORD, n * HEADS);
        k_fill_f32<<<cdiv(n * HEADS, 256), 256, 0, stream>>>(SSUM, 0.f, n * HEADS);
        k_edge_alpha<<<cdiv(E * HEADS, 256), 256, 0, stream>>>(Q, Kb, EP, src, dst, ALP, (unsigned*)MEN, E);
        k_edge_exp<<<cdiv(E * HEADS, 256), 256, 0, stream>>>(dst, ALP, (unsigned*)MEN, SSUM, E);
        k_edge_weight<<<cdiv(E * HEADS, 256), 256, 0, stream>>>(dst, ALP, SSUM, E);
        k_edge_aggr<<<cdiv(E * HCDIM, 256), 256, 0, stream>>>(src, dst, ALP, Vb, EP, Xout, E);
        k_leaky<<<cdiv(n * CDIM, 256), 256, 0, stream>>>(Xout, n * CDIM);
    };

    // ---- dense cross-attention conv (edge_attr == ones, fully connected) ----
    auto cross_attn = [&](const float* xq, int nq, const float* xkv, int nkv,
                          const TP& p, float* Xout) {
        k_cross_bias<<<cdiv(HCDIM, 256), 256, 0, stream>>>(p.We, p.bk, p.bv, B2K, B2V);
        gemm(stream, 0, xq,  CDIM, p.Wq, HCDIM, Q,  HCDIM, nq,  HCDIM, CDIM, 1.f, p.bq, 0, 0);
        gemm(stream, 0, xkv, CDIM, p.Wk, HCDIM, Kb, HCDIM, nkv, HCDIM, CDIM, 1.f, B2K, 0, 0);
        gemm(stream, 0, xkv, CDIM, p.Wv, HCDIM, Vb, HCDIM, nkv, HCDIM, CDIM, 1.f, B2V, 0, 0);
        for (int h = 0; h < HEADS; ++h)   // S_h = Q_h @ Kj_h^T / sqrt(C)
            gemm(stream, 1, Q + h * CDIM, HCDIM, Kb + h * CDIM, HCDIM,
                 S + (size_t)h * nq * nkv, nkv, nq, nkv, CDIM, invSqrtC, nullptr, 0, 0);
        k_softmax_row<<<HEADS * nq, 128, 0, stream>>>(S, nkv);
        gemm(stream, 0, xq, CDIM, p.Wskip, CDIM, Xout, CDIM, nq, CDIM, CDIM, 1.f, p.bskip, 0, 0);
        gemm(stream, 0, S, nkv, Vb, HCDIM, Xout, CDIM,
             nq, CDIM, nkv, 0.5f, nullptr, 1, 0);                       // head 0
        gemm(stream, 0, S + (size_t)nq * nkv, nkv, Vb + CDIM, HCDIM, Xout, CDIM,
             nq, CDIM, nkv, 0.5f, nullptr, 1, 1);                       // head 1 + leaky
    };

    // layer 0
    self_attn(x1, ei1, ea1, tsa, n1, E1, X1P);
    self_attn(x2, ei2, ea2, gsa, n2, E2, X2P);
    cross_attn(X1P, n1, X2P, n2, tca, X1PP);
    cross_attn(X2P, n2, X1P, n1, gca, X2PP);

    // pooling (written straight into d_out[0:600]) + MLP
    float* out = (float*)d_out;
    k_meancols<<<cdiv(CDIM, 256), 256, 0, stream>>>(X1PP, n1, out);
    k_meancols<<<cdiv(CDIM, 256), 256, 0, stream>>>(X2PP, n2, out + CDIM);
    k_vecmat<<<cdiv(HCDIM, 256), 256, 0, stream>>>(out, W1, b1, Y1, HCDIM, HCDIM, 1);
    k_vecmat<<<cdiv(CDIM, 256), 256, 0, stream>>>(Y1, W2, b2, Y2, HCDIM, CDIM, 1);
    k_final<<<1, 128, 0, stream>>>(Y2, W3, b3, out + 2 * CDIM);
}